// GNNBlock_88304527606467
// MI455X (gfx1250) — compile-verified
//
#include <hip/hip_runtime.h>
#include <math.h>

typedef __attribute__((ext_vector_type(2))) float v2f;
typedef __attribute__((ext_vector_type(8))) float v8f;

#define D 64
#define LN_EPS 1e-5f

// ---------------------------------------------------------------------------
// 1) deg[i] = 1.0 (self loop weight); also zero the LayerNorm accumulators
// ---------------------------------------------------------------------------
__global__ void init_deg_kernel(float* __restrict__ deg, double* __restrict__ sums, int N) {
  int i = blockIdx.x * blockDim.x + threadIdx.x;
  if (i < N) deg[i] = 1.0f;
  if (i == 0) { sums[0] = 0.0; sums[1] = 0.0; }
}

// ---------------------------------------------------------------------------
// 2) deg[dst[e]] += w[e]
// ---------------------------------------------------------------------------
__global__ void deg_accum_kernel(const int* __restrict__ dst, const float* __restrict__ w,
                                 float* __restrict__ deg, int E) {
  int e = blockIdx.x * blockDim.x + threadIdx.x;
  if (e < E) atomicAdd(&deg[dst[e]], w[e]);
}

// ---------------------------------------------------------------------------
// 3) dis[i] = deg[i] > 0 ? rsqrt(deg[i]) : 0   (in place)
// ---------------------------------------------------------------------------
__global__ void dis_kernel(float* __restrict__ deg, int N) {
  int i = blockIdx.x * blockDim.x + threadIdx.x;
  if (i < N) {
    float d = deg[i];
    deg[i] = (d > 0.0f) ? rsqrtf(d) : 0.0f;
  }
}

// ---------------------------------------------------------------------------
// 4) xw = x @ W^T  via V_WMMA_F32_16X16X4_F32.
//    Block = 256 threads = 8 waves; each wave computes a 16-row x 64-col tile.
//    f32 A (16x4) layout: lanes 0-15 hold {K=0,K=1}, lanes 16-31 hold {K=2,K=3}.
//    B (4x16) mirrored: lane l<16 holds B[0][l],B[1][l]; lane l+16 holds B[2][l],B[3][l],
//    where B[k][n] = W[n][k] (W is [D_OUT, D_IN] row-major).
//    C/D (16x16 f32): VGPR r, lanes 0-15 -> row r; lanes 16-31 -> row r+8.
// ---------------------------------------------------------------------------
__global__ __launch_bounds__(256) void gemm_xw_kernel(const float* __restrict__ x,
                                                      const float* __restrict__ W,
                                                      float* __restrict__ xw, int N) {
  __shared__ float sW[D * D];
  for (int i = threadIdx.x; i < D * D; i += 256) sW[i] = W[i];
  __syncthreads();

  const int lane  = threadIdx.x & 31;
  const int wave  = threadIdx.x >> 5;
  const int l16   = lane & 15;
  const int khalf = (lane >> 4) * 2;               // 0 for lanes 0-15, 2 for lanes 16-31
  const int row_base = (blockIdx.x * 8 + wave) * 16;
  if (row_base >= N) return;                        // wave-uniform exit (EXEC stays all-1s)
  const bool full_tile = (row_base + 16 <= N);      // wave-uniform

  // Preload A fragments for all 16 K-steps (32 f32 per lane).
  int arow = row_base + l16;
  if (arow >= N) arow = N - 1;                      // clamped load; store is guarded
  const float* xr = x + (size_t)arow * D;
  v2f afrag[16];
#pragma unroll
  for (int kk = 0; kk < 16; ++kk) {
    afrag[kk].x = xr[4 * kk + khalf + 0];
    afrag[kk].y = xr[4 * kk + khalf + 1];
  }

#pragma unroll
  for (int nt = 0; nt < 4; ++nt) {
    v8f acc = {};
    const float* wr = sW + (size_t)(nt * 16 + l16) * D + khalf;
#pragma unroll
    for (int kk = 0; kk < 16; ++kk) {
      v2f b;
      b.x = wr[4 * kk + 0];
      b.y = wr[4 * kk + 1];
      acc = __builtin_amdgcn_wmma_f32_16x16x4_f32(false, afrag[kk], false, b,
                                                  (short)0, acc, false, false);
    }
    const int col   = nt * 16 + l16;
    const int rbase = row_base + (lane >> 4) * 8;
    float* orow = xw + (size_t)rbase * D + col;
    if (full_tile) {
      // Uniform fast path: 8 unconditional stores with immediate offsets.
#pragma unroll
      for (int r = 0; r < 8; ++r) orow[(size_t)r * D] = acc[r];
    } else {
#pragma unroll
      for (int r = 0; r < 8; ++r) {
        if (rbase + r < N) orow[(size_t)r * D] = acc[r];
      }
    }
  }
}

// ---------------------------------------------------------------------------
// 5) out = xw * dis^2 (self-loop term: dis[i]*1*dis[i]) + bias
// ---------------------------------------------------------------------------
__global__ void init_out_kernel(const float* __restrict__ xw, const float* __restrict__ dis,
                                const float* __restrict__ b, float* __restrict__ out,
                                long long total) {
  long long i = (long long)blockIdx.x * blockDim.x + threadIdx.x;
  if (i >= total) return;
  int row = (int)(i >> 6);
  int col = (int)(i & 63);
  float s = dis[row];
  out[i] = xw[i] * s * s + b[col];
}

// ---------------------------------------------------------------------------
// 6) edge scatter: 16 threads per edge, each handles a float4 feature chunk.
//    out[dst] += xw[src] * (dis[src]*w*dis[dst])
// ---------------------------------------------------------------------------
__global__ __launch_bounds__(256) void edge_scatter_kernel(const int* __restrict__ src,
    const int* __restrict__ dstv, const float* __restrict__ w,
    const float* __restrict__ dis, const float* __restrict__ xw,
    float* __restrict__ out, int E) {
  long long t = (long long)blockIdx.x * blockDim.x + threadIdx.x;
  int e = (int)(t >> 4);
  int c = (int)(t & 15);
  if (e >= E) return;
  int s = src[e];
  int d = dstv[e];
  float norm = dis[s] * w[e] * dis[d];
  const float4 v = *reinterpret_cast<const float4*>(xw + (size_t)s * D + c * 4);
  float* o = out + (size_t)d * D + c * 4;
  atomicAdd(o + 0, v.x * norm);
  atomicAdd(o + 1, v.y * norm);
  atomicAdd(o + 2, v.z * norm);
  atomicAdd(o + 3, v.w * norm);
}

// ---------------------------------------------------------------------------
// 7a) global sum / sum-of-squares over all N*D entries (graph-mode LayerNorm)
// ---------------------------------------------------------------------------
__global__ __launch_bounds__(256) void reduce_kernel(const float* __restrict__ out,
                                                     double* __restrict__ sums,
                                                     long long total) {
  __shared__ double ssum[256];
  __shared__ double ssq[256];
  double ps = 0.0, pq = 0.0;
  for (long long i = (long long)blockIdx.x * 256 + threadIdx.x; i < total;
       i += (long long)gridDim.x * 256) {
    double v = (double)out[i];
    ps += v;
    pq += v * v;
  }
  ssum[threadIdx.x] = ps;
  ssq[threadIdx.x]  = pq;
  __syncthreads();
  for (int s = 128; s > 0; s >>= 1) {
    if ((int)threadIdx.x < s) {
      ssum[threadIdx.x] += ssum[threadIdx.x + s];
      ssq[threadIdx.x]  += ssq[threadIdx.x + s];
    }
    __syncthreads();
  }
  if (threadIdx.x == 0) {
    atomicAdd(&sums[0], ssum[0]);
    atomicAdd(&sums[1], ssq[0]);
  }
}

// ---------------------------------------------------------------------------
// 7b) normalize + affine + ReLU, in place on d_out
// ---------------------------------------------------------------------------
__global__ void finalize_kernel(float* __restrict__ out, const double* __restrict__ sums,
                                const float* __restrict__ gamma, const float* __restrict__ beta,
                                long long total) {
  long long i = (long long)blockIdx.x * blockDim.x + threadIdx.x;
  if (i >= total) return;
  double inv_n = 1.0 / (double)total;
  double mu    = sums[0] * inv_n;
  double var   = sums[1] * inv_n - mu * mu;
  float rstd   = rsqrtf((float)var + LN_EPS);
  int col = (int)(i & 63);
  float v = (out[i] - (float)mu) * rstd * gamma[col] + beta[col];
  out[i] = (v > 0.0f) ? v : 0.0f;
}

// ---------------------------------------------------------------------------
// launch
// ---------------------------------------------------------------------------
extern "C" void kernel_launch(void* const* d_in, const int* in_sizes, int n_in,
                              void* d_out, int out_size, void* d_ws, size_t ws_size,
                              hipStream_t stream) {
  const float* x     = (const float*)d_in[0];
  const int*   ei    = (const int*)d_in[1];
  const float* ew    = (const float*)d_in[2];
  const float* W     = (const float*)d_in[3];
  const float* b     = (const float*)d_in[4];
  const float* gamma = (const float*)d_in[5];
  const float* beta  = (const float*)d_in[6];

  const int N = in_sizes[0] / D;
  const int E = in_sizes[2];
  const int* src = ei;           // edge_index[0]
  const int* dst = ei + E;       // edge_index[1]

  // workspace layout: [dis: N f32][xw: N*64 f32][sums: 2 f64], 256B aligned
  char* ws = (char*)d_ws;
  float* dis = (float*)ws;
  size_t off = (((size_t)N * 4) + 255) & ~(size_t)255;
  float* xw = (float*)(ws + off);
  off += (((size_t)N * D * 4) + 255) & ~(size_t)255;
  double* sums = (double*)(ws + off);

  float* out = (float*)d_out;
  const long long total = (long long)N * D;

  init_deg_kernel<<<(N + 255) / 256, 256, 0, stream>>>(dis, sums, N);
  deg_accum_kernel<<<(E + 255) / 256, 256, 0, stream>>>(dst, ew, dis, E);
  dis_kernel<<<(N + 255) / 256, 256, 0, stream>>>(dis, N);
  gemm_xw_kernel<<<(N + 127) / 128, 256, 0, stream>>>(x, W, xw, N);
  init_out_kernel<<<(int)((total + 255) / 256), 256, 0, stream>>>(xw, dis, b, out, total);
  edge_scatter_kernel<<<(int)(((long long)E * 16 + 255) / 256), 256, 0, stream>>>(
      src, dst, ew, dis, xw, out, E);
  reduce_kernel<<<2048, 256, 0, stream>>>(out, sums, total);
  finalize_kernel<<<(int)((total + 255) / 256), 256, 0, stream>>>(out, sums, gamma, beta, total);
}